// KVCacheManager_55095840473791
// MI455X (gfx1250) — compile-verified
//
#include <hip/hip_runtime.h>

// Problem constants from the reference: L=2, B=8, H=8, S=4096, D=128
#define S_ 4096
// Total output floats: 2L*B*H*S*D = 134,217,728 -> 33,554,432 float4s (512 MB)

typedef float float4_t __attribute__((ext_vector_type(4)));

// Async DMA helpers. The 24-bit INST_OFFSET is applied to BOTH the LDS address
// (VDST/VSRC) and the global address (VADDR), so one base per phase suffices.
#define G2L(OFF)                                                            \
    asm volatile("global_load_async_to_lds_b128 %0, %1, off offset:" #OFF   \
                 " th:TH_LOAD_NT"                                           \
                 :: "v"(lds0), "v"(g) : "memory")
#define L2G(OFF)                                                            \
    asm volatile("global_store_async_from_lds_b128 %0, %1, off offset:" #OFF \
                 " th:TH_STORE_NT"                                          \
                 :: "v"(g), "v"(lds0) : "memory")
#define WAIT_ASYNC0() asm volatile("s_wait_asynccnt 0x0" ::: "memory")

// ---------------------------------------------------------------------------
// Bulk streaming copy: kv_cache -> out, staged through LDS with the gfx1250
// async DMA path. Global wave W owns 32 KB, processed in 4 iterations of 8 KB
// (16 async b128 instructions x 32 lanes x 16 B per phase).
// 2048 blocks x 8 waves x 32 KB = 512 MB exactly.
// ---------------------------------------------------------------------------
__global__ __launch_bounds__(256) void kv_copy_async(const float4_t* __restrict__ src,
                                                     float4_t* __restrict__ dst) {
    __shared__ float4_t buf[8 * 512];   // 8 KB staging per wave, 64 KB per block

    const int lane = threadIdx.x & 31;  // wave32
    const int wave = threadIdx.x >> 5;  // 8 waves per block
    float4_t* wbuf = buf + wave * 512;

    // LDS byte offset of this lane's staging slot (low 32 bits of the generic
    // pointer = as(3) offset; hardware adds the workgroup LDS_BASE).
    const unsigned lds0 = (unsigned)(unsigned long long)(const void*)(wbuf + lane);

    const long long W = (long long)blockIdx.x * 8 + wave;
    const float4_t* s0 = src + W * 2048 + lane;   // 2048 float4 = 32 KB per wave
    float4_t*       d0 = dst + W * 2048 + lane;

    #pragma unroll 1
    for (int it = 0; it < 4; ++it) {
        // Phase 1: global -> LDS, 16 x 512 B = 8 KB per wave outstanding.
        {
            unsigned long long g = (unsigned long long)(const void*)(s0 + it * 512);
            G2L(0);    G2L(512);  G2L(1024); G2L(1536);
            G2L(2048); G2L(2560); G2L(3072); G2L(3584);
            G2L(4096); G2L(4608); G2L(5120); G2L(5632);
            G2L(6144); G2L(6656); G2L(7168); G2L(7680);
        }
        WAIT_ASYNC0();   // LDS data valid

        // Phase 2: LDS -> global.
        {
            unsigned long long g = (unsigned long long)(void*)(d0 + it * 512);
            L2G(0);    L2G(512);  L2G(1024); L2G(1536);
            L2G(2048); L2G(2560); L2G(3072); L2G(3584);
            L2G(4096); L2G(4608); L2G(5120); L2G(5632);
            L2G(6144); L2G(6656); L2G(7168); L2G(7680);
        }
        WAIT_ASYNC0();   // LDS reusable next iteration
    }
    // s_endpgm performs an implicit wait-idle, so trailing stores complete.
}

// ---------------------------------------------------------------------------
// Scatter: overwrite row pos[b] of each (layer-kv, b, h) slice with new_kv.
// 256 rows x 32 float4 = 8192 threads (128 KB of writes).
// out float4 index = (((lidx*B + b)*H + h)*S + pos[b]) * 32 + d4
// new_kv float4 index = t (contiguous)
// ---------------------------------------------------------------------------
__global__ void kv_scatter(const float4_t* __restrict__ newkv,
                           const int* __restrict__ pos,
                           float4_t* __restrict__ out) {
    const int t   = blockIdx.x * blockDim.x + threadIdx.x;  // 0..8191
    const int d4  = t & 31;
    const int row = t >> 5;                                 // ((lidx*B + b)*H + h)
    const int b   = (row >> 3) & 7;

    const long long p      = (long long)pos[b];             // 0..S-1
    const long long dstIdx = ((long long)row * S_ + p) * 32 + d4;
    out[dstIdx] = newkv[t];
}

extern "C" void kernel_launch(void* const* d_in, const int* in_sizes, int n_in,
                              void* d_out, int out_size, void* d_ws, size_t ws_size,
                              hipStream_t stream) {
    const float4_t* kv  = (const float4_t*)d_in[0];  // (2L,B,H,S,D) f32
    const float4_t* nkv = (const float4_t*)d_in[1];  // (L,2,B,H,1,D) f32
    const int*      pos = (const int*)d_in[2];       // (B,1) int

    float4_t* out = (float4_t*)d_out;                // (2L,B,H,S,D) f32

    // 1) Full-cache streaming copy via async LDS DMA (512 MB).
    kv_copy_async<<<2048, 256, 0, stream>>>(kv, out);
    // 2) Tiny scatter of the new tokens (stream-ordered after the copy).
    kv_scatter<<<32, 256, 0, stream>>>(nkv, pos, out);
}